// Complex_CA_20040317403312
// MI455X (gfx1250) — compile-verified
//
#include <hip/hip_runtime.h>
#include <hip/hip_bf16.h>

typedef float v2f __attribute__((ext_vector_type(2)));
typedef float v8f __attribute__((ext_vector_type(8)));

#define W 17
#define NPIX 289          // 17*17
#define PS 20             // padded row stride
#define PH 19             // padded rows (1-px zero halo)
#define PSZ (PH*PS)       // 380
#define NT 19             // pixel tiles of 16
#define TP 304            // NT*16

__device__ __forceinline__ float clip10(float v) { return fminf(fmaxf(v, -10.f), 10.f); }

__global__ __launch_bounds__(512)
void ca_persistent_kernel(const float* __restrict__ cell_in,
                          const float* __restrict__ food,
                          const float* __restrict__ w3,
                          const float* __restrict__ b3,
                          const float* __restrict__ w4,
                          const float* __restrict__ b4,
                          const int*   __restrict__ steps_p,
                          float* __restrict__ out)
{
    __shared__ float bufL[2][3][PSZ];   // dynamic channels 0..2, double buffered
    __shared__ float scentP[PSZ];       // constant channel 3 (scent), padded
    __shared__ float yL[12][TP];        // perception matrix (12 x pixels)
    __shared__ float hL[16][TP];        // hidden layer (16 x pixels)
    __shared__ float preA[TP];
    __shared__ float maskL[TP];
    __shared__ float b3L[16];
    __shared__ float b4L[4];

    const int tid  = threadIdx.x;
    const int lane = tid & 31;
    // Force the wave index scalar so the WMMA tile loops compile as uniform
    // (s_cmp/s_cbranch) with EXEC untouched — ISA requires EXEC all-1s for WMMA.
    const int wv   = __builtin_amdgcn_readfirstlane(tid >> 5);   // 16 waves
    const int m    = lane & 15;
    const int hi   = lane >> 4;
    const int steps = steps_p[0];

    // Loop-invariant per-thread indices (hoisted out of the 256-step loop)
    const bool act = (tid < NPIX);
    const int  pi  = tid / W;                 // pixel row
    const int  pj  = tid - pi * W;            // pixel col
    const int  q   = (pi + 1) * PS + (pj + 1);// padded LDS index for this pixel

    // ---- zero-init LDS (halos must be 0 and stay 0) ----
    for (int i = tid; i < 2*3*PSZ; i += 512) ((float*)bufL)[i] = 0.f;
    for (int i = tid; i < PSZ;     i += 512) scentP[i] = 0.f;
    for (int i = tid; i < 12*TP;   i += 512) ((float*)yL)[i] = 0.f;
    for (int i = tid; i < 16*TP;   i += 512) ((float*)hL)[i] = 0.f;
    if (tid < 16) b3L[tid] = b3[tid];
    if (tid < 4)  b4L[tid] = b4[tid];

    // 5x5 scent kernel (cross-correlation, zero SAME padding) — food is constant
    const float K5[5][5] = {
        {0.f,   0.125f, 0.25f, 0.125f, 0.f},
        {0.125f,0.25f,  0.5f,  0.25f,  0.125f},
        {0.25f, 0.5f,   1.0f,  0.5f,   0.25f},
        {0.125f,0.25f,  0.5f,  0.25f,  0.125f},
        {0.f,   0.125f, 0.25f, 0.125f, 0.f}};

    if (act) {
        float s = 0.f;
        #pragma unroll
        for (int a = 0; a < 5; a++)
            #pragma unroll
            for (int b = 0; b < 5; b++) {
                const int ii = pi + a - 2, jj = pj + b - 2;
                if (ii >= 0 && ii < W && jj >= 0 && jj < W) s += K5[a][b] * food[ii*W + jj];
            }
        scentP[q] = s;
        for (int c = 0; c < 3; c++)
            bufL[0][c][q] = cell_in[c*NPIX + tid];
    }
    __syncthreads();

    // Static perception rows for channel 3 (cell[3] == scent every step)
    if (act) {
        yL[3][tid] = scentP[q];
        yL[7][tid] =                                    // DX (Sobel/8)
            -0.125f*scentP[q-PS-1] + 0.125f*scentP[q-PS+1]
            -0.25f *scentP[q-1]    + 0.25f *scentP[q+1]
            -0.125f*scentP[q+PS-1] + 0.125f*scentP[q+PS+1];
        yL[11][tid] =                                   // DY
            -0.125f*scentP[q-PS-1] - 0.25f*scentP[q-PS] - 0.125f*scentP[q-PS+1]
            +0.125f*scentP[q+PS-1] + 0.25f*scentP[q+PS] + 0.125f*scentP[q+PS+1];
    }

    // ---- A-matrix fragments (weights) in registers, per documented 16x4 f32 layout ----
    // A: lane(0-15)->M, lane(16-31)->M-16; VGPR0 holds K {0|2}, VGPR1 holds K {1|3}
    v2f a1[3], a2[4];
    #pragma unroll
    for (int kc = 0; kc < 3; kc++) {
        const int k0 = kc*4 + (hi ? 2 : 0);
        a1[kc].x = w3[m*12 + k0];
        a1[kc].y = w3[m*12 + k0 + 1];
    }
    #pragma unroll
    for (int kc = 0; kc < 4; kc++) {
        const int k0 = kc*4 + (hi ? 2 : 0);
        a2[kc].x = (m < 4) ? w4[m*16 + k0]     : 0.f;
        a2[kc].y = (m < 4) ? w4[m*16 + k0 + 1] : 0.f;
    }
    __syncthreads();

    for (int s = 0; s < steps; s++) {
        float (*cur)[PSZ] = bufL[s & 1];
        float (*nxt)[PSZ] = bufL[(s + 1) & 1];

        // ---- Phase 1: pre_alive + perception rows for dynamic channels ----
        if (act) {
            float mx = cur[0][q-PS-1];
            mx = fmaxf(mx, cur[0][q-PS]);   mx = fmaxf(mx, cur[0][q-PS+1]);
            mx = fmaxf(mx, cur[0][q-1]);    mx = fmaxf(mx, cur[0][q]);
            mx = fmaxf(mx, cur[0][q+1]);    mx = fmaxf(mx, cur[0][q+PS-1]);
            mx = fmaxf(mx, cur[0][q+PS]);   mx = fmaxf(mx, cur[0][q+PS+1]);
            preA[tid] = (mx > 0.1f) ? 1.f : 0.f;
            #pragma unroll
            for (int c = 0; c < 3; c++) {
                const float* P = cur[c];
                yL[c][tid] = P[q];
                yL[4+c][tid] =
                    -0.125f*P[q-PS-1] + 0.125f*P[q-PS+1]
                    -0.25f *P[q-1]    + 0.25f *P[q+1]
                    -0.125f*P[q+PS-1] + 0.125f*P[q+PS+1];
                yL[8+c][tid] =
                    -0.125f*P[q-PS-1] - 0.25f*P[q-PS] - 0.125f*P[q-PS+1]
                    +0.125f*P[q+PS-1] + 0.25f*P[q+PS] + 0.125f*P[q+PS+1];
            }
        }
        __syncthreads();

        // ---- Phase 2: layer 1  h = relu(W3 @ y + b3)  via V_WMMA_F32_16X16X4_F32 ----
        for (int t = wv; t < NT; t += 16) {           // scalar-uniform trip count
            const int p = t*16 + m;
            v8f acc = {};
            #pragma unroll
            for (int kc = 0; kc < 3; kc++) {          // K = 12 in chunks of 4
                const int k0 = kc*4 + (hi ? 2 : 0);
                v2f b; b.x = yL[k0][p]; b.y = yL[k0+1][p];
                acc = __builtin_amdgcn_wmma_f32_16x16x4_f32(
                          false, a1[kc], false, b, (short)0, acc, false, false);
            }
            const int Mb = hi ? 8 : 0;                // C/D layout: VGPR r -> M = Mb + r
            #pragma unroll
            for (int r = 0; r < 8; r++)
                hL[Mb + r][p] = fmaxf(acc[r] + b3L[Mb + r], 0.f);
        }
        __syncthreads();

        // ---- Phase 3: layer 2  delta = W4 @ h + b4  via WMMA; write pre-mask x ----
        for (int t = wv; t < NT; t += 16) {
            const int p = t*16 + m;
            v8f acc = {};
            #pragma unroll
            for (int kc = 0; kc < 4; kc++) {          // K = 16 in chunks of 4
                const int k0 = kc*4 + (hi ? 2 : 0);
                v2f b; b.x = hL[k0][p]; b.y = hL[k0+1][p];
                acc = __builtin_amdgcn_wmma_f32_16x16x4_f32(
                          false, a2[kc], false, b, (short)0, acc, false, false);
            }
            if (hi == 0 && p < NPIX) {                // rows M=0..2 live in lanes 0-15
                const int ri = p / W, rj = p - ri * W;
                const int rq = (ri + 1) * PS + (rj + 1);
                #pragma unroll
                for (int c = 0; c < 3; c++)
                    nxt[c][rq] = cur[c][rq] + acc[c] + b4L[c];
            }
        }
        __syncthreads();

        // ---- Phase 4a: post_alive & mask ----
        if (act) {
            float mx = nxt[0][q-PS-1];
            mx = fmaxf(mx, nxt[0][q-PS]);   mx = fmaxf(mx, nxt[0][q-PS+1]);
            mx = fmaxf(mx, nxt[0][q-1]);    mx = fmaxf(mx, nxt[0][q]);
            mx = fmaxf(mx, nxt[0][q+1]);    mx = fmaxf(mx, nxt[0][q+PS-1]);
            mx = fmaxf(mx, nxt[0][q+PS]);   mx = fmaxf(mx, nxt[0][q+PS+1]);
            maskL[tid] = (preA[tid] != 0.f && mx > 0.1f) ? 1.f : 0.f;
        }
        __syncthreads();

        // ---- Phase 4b: apply mask + clip ----
        if (act) {
            const float mk = maskL[tid];
            #pragma unroll
            for (int c = 0; c < 3; c++)
                nxt[c][q] = clip10(nxt[c][q] * mk);
        }
        __syncthreads();
    }

    // ---- Outputs: cell(4x289) | food(289) | living_count(1) ----
    float (*fin)[PSZ] = bufL[steps & 1];
    if (act) {
        for (int c = 0; c < 3; c++) out[c*NPIX + tid] = fin[c][q];
        out[3*NPIX + tid] = clip10(scentP[q]);   // channel 3 restored to scent
        out[4*NPIX + tid] = food[tid];
    }
    __syncthreads();
    if (tid == 0) {                               // deterministic serial reduction
        float s = 0.f;
        for (int p = 0; p < NPIX; p++) {
            const int i = p / W, j = p - (p / W) * W;
            s += fin[0][(i+1)*PS + (j+1)];
        }
        out[5*NPIX] = s;                          // index 1445
    }
}

extern "C" void kernel_launch(void* const* d_in, const int* in_sizes, int n_in,
                              void* d_out, int out_size, void* d_ws, size_t ws_size,
                              hipStream_t stream) {
    (void)in_sizes; (void)n_in; (void)out_size; (void)d_ws; (void)ws_size;
    const float* cell  = (const float*)d_in[0];
    const float* food  = (const float*)d_in[1];
    const float* w3    = (const float*)d_in[2];
    const float* b3    = (const float*)d_in[3];
    const float* w4    = (const float*)d_in[4];
    const float* b4    = (const float*)d_in[5];
    const int*   steps = (const int*)  d_in[6];
    float* out = (float*)d_out;
    // Single persistent workgroup: the 256-step chain is strictly sequential;
    // all state lives in LDS, WMMA does the dense layers.
    ca_persistent_kernel<<<1, 512, 0, stream>>>(cell, food, w3, b3, w4, b4, steps, out);
}